// ClassWiseCrossEntropyLoss_62646392979576
// MI455X (gfx1250) — compile-verified
//
#include <hip/hip_runtime.h>
#include <hip/hip_bf16.h>

typedef __attribute__((ext_vector_type(2))) float v2f;
typedef __attribute__((ext_vector_type(8))) float v8f;

namespace {
constexpr int kB = 2;
constexpr int kC = 16;
constexpr int kS = 64 * 128 * 128;      // voxels per batch = 1,048,576 (2^20)
constexpr int kN4 = (kB * kS) / 4;      // 524,288 groups of 4 voxels
}

// ---------------------------------------------------------------------------
// Zero the 64-dword accumulator workspace (32 f32 sums + 32 u32 counts).
// d_ws is poisoned by the harness and never re-zeroed between replays.
// ---------------------------------------------------------------------------
__global__ void cwce_zero_ws(unsigned int* __restrict__ ws) {
  if (threadIdx.x < 64) ws[threadIdx.x] = 0u;
}

// ---------------------------------------------------------------------------
// Streaming pass: log-softmax over C=16, gather label logit, bin NLL into
// 32 (b,class) segments. 4 voxels/thread -> all x reads are b128 coalesced.
// LDS atomics for intra-block binning, 64 global atomics per block to flush.
// ---------------------------------------------------------------------------
__global__ __launch_bounds__(256) void cwce_main(
    const float* __restrict__ x, const int* __restrict__ y,
    float* __restrict__ gsum, unsigned int* __restrict__ gcnt) {
  __shared__ float lsum[32];
  __shared__ unsigned int lcnt[32];
  const int tid = threadIdx.x;
  if (tid < 32) { lsum[tid] = 0.0f; lcnt[tid] = 0u; }
  __syncthreads();

  const int stride = gridDim.x * blockDim.x;
  for (int n4 = blockIdx.x * blockDim.x + tid; n4 < kN4; n4 += stride) {
    const int b = n4 >> 18;                       // (n4*4) / 2^20
    const int s = (n4 & ((kS / 4) - 1)) << 2;     // spatial offset, mult of 4
    const float* xb = x + (size_t)b * ((size_t)kC * kS) + s;

    // speculative prefetch of the next class plane (global_prefetch_b8)
    __builtin_prefetch(xb + kS, 0, 0);

    float4 l[kC];
#pragma unroll
    for (int c = 0; c < kC; ++c)
      l[c] = *(const float4*)(xb + (size_t)c * kS);

    const int4 yv = *(const int4*)(y + (size_t)b * kS + s);

    float4 mx = l[0];
#pragma unroll
    for (int c = 1; c < kC; ++c) {
      mx.x = fmaxf(mx.x, l[c].x);
      mx.y = fmaxf(mx.y, l[c].y);
      mx.z = fmaxf(mx.z, l[c].z);
      mx.w = fmaxf(mx.w, l[c].w);
    }

    float p0 = 0.f, p1 = 0.f, p2 = 0.f, p3 = 0.f;
    float e0 = 0.f, e1 = 0.f, e2 = 0.f, e3 = 0.f;
#pragma unroll
    for (int c = 0; c < kC; ++c) {
      e0 += __expf(l[c].x - mx.x);
      e1 += __expf(l[c].y - mx.y);
      e2 += __expf(l[c].z - mx.z);
      e3 += __expf(l[c].w - mx.w);
      p0 = (c == yv.x) ? l[c].x : p0;   // v_cndmask select of labeled logit
      p1 = (c == yv.y) ? l[c].y : p1;
      p2 = (c == yv.z) ? l[c].z : p2;
      p3 = (c == yv.w) ? l[c].w : p3;
    }

    const float n0 = (__logf(e0) + mx.x) - p0;
    const float n1 = (__logf(e1) + mx.y) - p1;
    const float n2 = (__logf(e2) + mx.z) - p2;
    const float n3 = (__logf(e3) + mx.w) - p3;

    const unsigned int segbase = (unsigned int)b * 16u;
    atomicAdd(&lsum[segbase + (unsigned int)yv.x], n0);
    atomicAdd(&lcnt[segbase + (unsigned int)yv.x], 1u);
    atomicAdd(&lsum[segbase + (unsigned int)yv.y], n1);
    atomicAdd(&lcnt[segbase + (unsigned int)yv.y], 1u);
    atomicAdd(&lsum[segbase + (unsigned int)yv.z], n2);
    atomicAdd(&lcnt[segbase + (unsigned int)yv.z], 1u);
    atomicAdd(&lsum[segbase + (unsigned int)yv.w], n3);
    atomicAdd(&lcnt[segbase + (unsigned int)yv.w], 1u);
  }

  __syncthreads();
  if (tid < 32) {
    atomicAdd(&gsum[tid], lsum[tid]);
    atomicAdd(&gcnt[tid], lcnt[tid]);
  }
}

// ---------------------------------------------------------------------------
// Finalize (1 wave): per-segment means, then 32->1 reduction via a real
// v_wmma_f32_16x16x4_f32 against an all-ones B matrix.
// A layout (16x4 f32): lanes 0-15 VGPR0=K0/VGPR1=K1, lanes 16-31 VGPR0=K2/
// VGPR1=K3, M = lane%16. So a[0]=mean(lane), a[1]=0 places all 32 means.
// D[m,n] = mean_m + mean_{m+16} (same for every n). Lanes 0..15 hold rows
// 0..7, lanes 16..31 hold rows 8..15 -> sum 8 acc VGPRs, combine lanes 0&16.
// ---------------------------------------------------------------------------
__global__ void cwce_final(const float* __restrict__ gsum,
                           const unsigned int* __restrict__ gcnt,
                           float* __restrict__ out) {
  const int lane = threadIdx.x;  // 0..31, single wave, EXEC all-ones
  const float s = gsum[lane];
  const unsigned int c = gcnt[lane];
  const float mean = (c != 0u) ? (s / (float)c) : 0.0f;

  v2f a;   a[0] = mean; a[1] = 0.0f;
  v2f one; one[0] = 1.0f; one[1] = 1.0f;
  v8f acc = {};
  acc = __builtin_amdgcn_wmma_f32_16x16x4_f32(
      /*neg_a=*/false, a, /*neg_b=*/false, one,
      /*c_mod=*/(short)0, acc, /*reuse_a=*/false, /*reuse_b=*/false);

  float part = 0.0f;
#pragma unroll
  for (int i = 0; i < 8; ++i) part += acc[i];

  const float tot = __shfl(part, 0, 32) + __shfl(part, 16, 32);
  if (lane == 0) out[0] = tot * (1.0f / 32.0f);   // / (B*C)
}

// ---------------------------------------------------------------------------
extern "C" void kernel_launch(void* const* d_in, const int* in_sizes, int n_in,
                              void* d_out, int out_size, void* d_ws, size_t ws_size,
                              hipStream_t stream) {
  (void)in_sizes; (void)n_in; (void)out_size; (void)ws_size;
  const float* x = (const float*)d_in[0];
  const int* y = (const int*)d_in[1];
  float* gsum = (float*)d_ws;
  unsigned int* gcnt = (unsigned int*)d_ws + 32;

  cwce_zero_ws<<<1, 64, 0, stream>>>((unsigned int*)d_ws);
  // 2048 blocks x 256 threads = 524,288 threads: exactly one float4 group each
  cwce_main<<<2048, 256, 0, stream>>>(x, y, gsum, gcnt);
  cwce_final<<<1, 32, 0, stream>>>(gsum, gcnt, (float*)d_out);
}